// RandomProjectionModule_16355235463553
// MI455X (gfx1250) — compile-verified
//
#include <hip/hip_runtime.h>
#include <stdint.h>

#define TDW 1e-4f
#define NODES 100000
#define DIM 128
#define TILE_ROWS 80   // 100000 / 80 = 1250 tiles exactly; 80*128*4 = 40 KB LDS

typedef int v4i_ __attribute__((ext_vector_type(4)));
typedef int v8i_ __attribute__((ext_vector_type(8)));

// ---------------------------------------------------------------------------
// Kernel 1: out[0] = proj[0] (facs[0] == 1 exactly) — pure copy done with the
// CDNA5 Tensor Data Mover: global -> LDS -> global, one wave per 80x128 tile.
// D# layout per CDNA5 ISA ch.8 (group0: count/lds_addr/global_addr/type=2;
// group1: data_size=4B, tensor 128 x 80, tile 128 x 80, stride0 = 128).
// ---------------------------------------------------------------------------
__global__ __launch_bounds__(32) void copy_layer0_tdm(const float* __restrict__ src,
                                                      float* __restrict__ dst) {
  __shared__ __align__(16) float tile[TILE_ROWS * DIM];
  const uint64_t elems  = (uint64_t)blockIdx.x * (TILE_ROWS * DIM);
  const uint64_t ga_in  = (uint64_t)(uintptr_t)(src + elems);
  const uint64_t ga_out = (uint64_t)(uintptr_t)(dst + elems);
  const uint32_t lds    = (uint32_t)(uintptr_t)&tile[0];   // low 32 bits = LDS offset

  // D# group 0 (4 SGPRs): count=1 | lds_addr | global_addr[31:0] | addr[56:32], type=2
  v4i_ g0l = { (int)1u, (int)lds, (int)(uint32_t)ga_in,
               (int)(((uint32_t)(ga_in >> 32) & 0x01FFFFFFu) | (2u << 30)) };
  v4i_ g0s = { (int)1u, (int)lds, (int)(uint32_t)ga_out,
               (int)(((uint32_t)(ga_out >> 32) & 0x01FFFFFFu) | (2u << 30)) };
  // D# group 1 (8 SGPRs):
  //  d0: wg_mask=0, data_size=2 (4B), no barrier/iterate/pad
  //  bits[79:48]  tensor_dim0 = 128
  //  bits[111:80] tensor_dim1 = 80
  //  bits[127:112] tile_dim0  = 128
  //  bits[143:128] tile_dim1  = 80 ; tile_dim2 = 0 (2-D)
  //  bits[207:160] tensor_dim0_stride = 128 ; dim1_stride unused
  v8i_ g1 = { (int)(2u << 16),
              (int)((DIM & 0xFFFFu) << 16),
              (int)(((unsigned)DIM >> 16) | ((TILE_ROWS & 0xFFFFu) << 16)),
              (int)(((unsigned)TILE_ROWS >> 16) | ((unsigned)DIM << 16)),
              (int)(TILE_ROWS),
              (int)(DIM),
              0, 0 };
  v4i_ gz = { 0, 0, 0, 0 };  // groups 2/3: zeros (dims beyond 2-D unused)

  asm volatile("tensor_load_to_lds %0, %1, %2, %3"
               :: "s"(g0l), "s"(g1), "s"(gz), "s"(gz) : "memory");
  __builtin_amdgcn_s_wait_tensorcnt(0);
  asm volatile("tensor_store_from_lds %0, %1, %2, %3"
               :: "s"(g0s), "s"(g1), "s"(gz), "s"(gz) : "memory");
  __builtin_amdgcn_s_wait_tensorcnt(0);
}

// ---------------------------------------------------------------------------
// Kernel 2: out[i] = proj[i] * decay^i for i = 1..3 (blockIdx.y = i-1).
// float4 streaming; uniform exp computed once per block, broadcast via LDS.
// ---------------------------------------------------------------------------
__global__ __launch_bounds__(256) void scale_layers(const float4* __restrict__ proj,
                                                    float4* __restrict__ out,
                                                    const float* __restrict__ times,
                                                    const float* __restrict__ now,
                                                    int E) {
  __shared__ float s_fac;
  const int layer = blockIdx.y + 1;
  if (threadIdx.x == 0) {
    float tN  = times[E - 1];
    float dec = __expf(-TDW * (tN - now[0]));
    float f = dec;
    for (int k = 1; k < layer; ++k) f *= dec;
    s_fac = f;
  }
  __syncthreads();
  const int ND4 = NODES * DIM / 4;
  int i = blockIdx.x * blockDim.x + threadIdx.x;
  if (i < ND4) {
    float4 v = proj[(size_t)layer * ND4 + i];
    float f = s_fac;
    v.x *= f; v.y *= f; v.z *= f; v.w *= f;
    out[(size_t)layer * ND4 + i] = v;
  }
}

// ---------------------------------------------------------------------------
// Kernel 3: edge scatter. One wave32 per (layer, edge): lanes cover the 128-f32
// row as float4 (512 B fully coalesced). Gathers from PRE-update proj[i-1],
// scales by facs[i-1] * exp(-w*(tN - t[e])), atomically adds into both the
// src and dst rows of out[i].
// ---------------------------------------------------------------------------
__global__ __launch_bounds__(256) void scatter_edges(const int* __restrict__ srcid,
                                                     const int* __restrict__ dstid,
                                                     const float* __restrict__ times,
                                                     const float* __restrict__ proj,
                                                     float* __restrict__ out,
                                                     const float* __restrict__ now,
                                                     int E) {
  __shared__ float sh_tN, sh_fac;
  const int layer = blockIdx.y + 1;          // 1..3
  if (threadIdx.x == 0) {
    float tN  = times[E - 1];
    float dec = __expf(-TDW * (tN - now[0]));
    float f = 1.0f;                          // facs[layer-1] = dec^(layer-1)
    for (int k = 1; k < layer; ++k) f *= dec;
    sh_tN = tN; sh_fac = f;
  }
  __syncthreads();

  const int wave = threadIdx.x >> 5;
  const int lane = threadIdx.x & 31;
  const int e = blockIdx.x * 8 + wave;
  if (e >= E) return;

  const int s = srcid[e];
  const int d = dstid[e];
  const float tw   = __expf(-TDW * (sh_tN - times[e]));
  const float coef = sh_fac * tw;

  const float4* rowFromD = (const float4*)(proj + ((size_t)(layer - 1) * NODES + d) * DIM);
  const float4* rowFromS = (const float4*)(proj + ((size_t)(layer - 1) * NODES + s) * DIM);
  float* outS = out + ((size_t)layer * NODES + s) * DIM + (size_t)lane * 4;
  float* outD = out + ((size_t)layer * NODES + d) * DIM + (size_t)lane * 4;

  __builtin_prefetch(outS, 1, 0);            // global_prefetch of atomic targets
  __builtin_prefetch(outD, 1, 0);

  float4 a = rowFromD[lane];                 // message into src row
  float4 b = rowFromS[lane];                 // message into dst row

  atomicAdd(outS + 0, a.x * coef);
  atomicAdd(outS + 1, a.y * coef);
  atomicAdd(outS + 2, a.z * coef);
  atomicAdd(outS + 3, a.w * coef);
  atomicAdd(outD + 0, b.x * coef);
  atomicAdd(outD + 1, b.y * coef);
  atomicAdd(outD + 2, b.z * coef);
  atomicAdd(outD + 3, b.w * coef);
}

// ---------------------------------------------------------------------------
extern "C" void kernel_launch(void* const* d_in, const int* in_sizes, int n_in,
                              void* d_out, int out_size, void* d_ws, size_t ws_size,
                              hipStream_t stream) {
  const int*   srcid = (const int*)d_in[0];
  const int*   dstid = (const int*)d_in[1];
  const float* times = (const float*)d_in[2];
  const float* proj  = (const float*)d_in[3];
  const float* now   = (const float*)d_in[4];
  float* out = (float*)d_out;
  const int E = in_sizes[0];

  // Layer 0: TDM tile copy (facs[0] == 1).
  copy_layer0_tdm<<<dim3(NODES / TILE_ROWS), dim3(32), 0, stream>>>(proj, out);

  // Layers 1..3: base scaling.
  const int ND4 = NODES * DIM / 4;
  scale_layers<<<dim3((ND4 + 255) / 256, 3), dim3(256), 0, stream>>>(
      (const float4*)proj, (float4*)out, times, now, E);

  // Layers 1..3: edge scatter-add (must follow the base writes in stream order).
  scatter_edges<<<dim3((E + 7) / 8, 3), dim3(256), 0, stream>>>(
      srcid, dstid, times, proj, out, now, E);
}